// RelationLayer_6442450944302
// MI455X (gfx1250) — compile-verified
//
#include <hip/hip_runtime.h>
#include <math.h>

#define BB 8
#define LL 256
#define DD 400

typedef __attribute__((ext_vector_type(2))) float v2f;
typedef __attribute__((ext_vector_type(8))) float v8f;

// ---------------------------------------------------------------------------
// Kernel 1: g0 = x @ W0^T + b0, g1 = x @ W1^T + b1 via V_WMMA_F32_16X16X4_F32.
// One wave (32 lanes) per 16x16 output tile; A operand (x tile) shared between
// the two WMMA accumulators. grid = (DD/16, LL/16, BB), block = 32.
//
// A (16x4 f32, ISA 7.12.2): lanes 0-15 hold M=lane, {K=k+0,k+1}; lanes 16-31
// hold M=lane-16, {K=k+2,k+3}.  B (4x16) mirrors with N across lanes.
// C/D (16x16 f32): VGPR r -> M=r (lanes 0-15) / M=r+8 (lanes 16-31), N=lane&15.
// ---------------------------------------------------------------------------
__global__ void relation_gemm_wmma(const float* __restrict__ x,
                                   const float* __restrict__ W0,
                                   const float* __restrict__ b0v,
                                   const float* __restrict__ W1,
                                   const float* __restrict__ b1v,
                                   float* __restrict__ g0,
                                   float* __restrict__ g1) {
    const int n0   = blockIdx.x * 16;    // output feature tile (e)
    const int i0   = blockIdx.y * 16;    // sequence tile (l)
    const int b    = blockIdx.z;         // batch
    const int lane = threadIdx.x;        // 0..31
    const int rc   = lane & 15;          // A row / B col within tile
    const int koff = (lane >> 4) << 1;   // 0 or 2

    const float* xrow  = x  + ((size_t)b * LL + (size_t)(i0 + rc)) * DD;
    const float* w0row = W0 + (size_t)(n0 + rc) * DD;   // g[l,e] = sum_d x[l,d]*W[e,d]
    const float* w1row = W1 + (size_t)(n0 + rc) * DD;

    v8f c0 = {};
    v8f c1 = {};
    for (int k = 0; k < DD; k += 4) {
        v2f a  = *(const v2f*)(xrow  + k + koff);
        v2f w0 = *(const v2f*)(w0row + k + koff);
        v2f w1 = *(const v2f*)(w1row + k + koff);
        c0 = __builtin_amdgcn_wmma_f32_16x16x4_f32(false, a, false, w0,
                                                   (short)0, c0, false, false);
        c1 = __builtin_amdgcn_wmma_f32_16x16x4_f32(false, a, false, w1,
                                                   (short)0, c1, false, false);
    }

    const float bias0 = b0v[n0 + rc];
    const float bias1 = b1v[n0 + rc];
    const int rowbase = i0 + ((lane < 16) ? 0 : 8);
    float* g0p = g0 + (size_t)b * LL * DD + n0 + rc;
    float* g1p = g1 + (size_t)b * LL * DD + n0 + rc;
#pragma unroll
    for (int r = 0; r < 8; ++r) {
        const size_t off = (size_t)(rowbase + r) * DD;
        g0p[off] = c0[r] + bias0;
        g1p[off] = c1[r] + bias1;
    }
}

// ---------------------------------------------------------------------------
// Kernel 2: pairwise gate (compute-bound core).
// Block = (d, b); 256 threads (thread = i). Stage x[b,:,d]*mask and g1[b,:,d]
// in LDS, then each thread accumulates over j with sigmoid via exp2+rcp trans
// ops. 2 trans + ~4 VALU + 2 LDS reads per gate element; 210M elements total.
// ---------------------------------------------------------------------------
__global__ void relation_gate_kernel(const float* __restrict__ x,
                                     const int* __restrict__ mask,
                                     const float* __restrict__ g0,
                                     const float* __restrict__ g1,
                                     float* __restrict__ out) {
    const int d = blockIdx.x;   // 0..DD-1
    const int b = blockIdx.y;   // 0..BB-1
    const int i = threadIdx.x;  // 0..LL-1

    __shared__ float xs[LL];    // x[b,j,d] * mask[b,j]
    __shared__ float g1s[LL];   // g1[b,j,d]
    __shared__ float slen;

    const float m = (float)mask[b * LL + i];
    if (i == 0) slen = 0.0f;
    __syncthreads();
    atomicAdd(&slen, m);                                  // ds_add_f32
    xs[i]  = x [((size_t)b * LL + i) * DD + d] * m;
    g1s[i] = g1[((size_t)b * LL + i) * DD + d];
    __syncthreads();

    const float len     = slen;
    const float inv_len = (len > 0.0f) ? __builtin_amdgcn_rcpf(len) : 0.0f;
    const float a       = g0[((size_t)b * LL + i) * DD + d];
    const float NEG_LOG2E = -1.4426950408889634f;

    float acc = 0.0f;
#pragma unroll 8
    for (int j = 0; j < LL; ++j) {
        const float z = a + g1s[j];
        const float e = __builtin_amdgcn_exp2f(z * NEG_LOG2E); // exp(-z)
        const float s = __builtin_amdgcn_rcpf(1.0f + e);       // sigmoid(z)
        acc += s * xs[j];
    }

    out[((size_t)b * LL + i) * DD + d] = tanhf(acc * inv_len * m);
}

// ---------------------------------------------------------------------------
extern "C" void kernel_launch(void* const* d_in, const int* in_sizes, int n_in,
                              void* d_out, int out_size, void* d_ws, size_t ws_size,
                              hipStream_t stream) {
    const float* x    = (const float*)d_in[0];
    const int*   mask = (const int*)  d_in[1];
    const float* W0   = (const float*)d_in[2];
    const float* b0v  = (const float*)d_in[3];
    const float* W1   = (const float*)d_in[4];
    const float* b1v  = (const float*)d_in[5];
    float*       out  = (float*)d_out;

    float* g0 = (float*)d_ws;
    float* g1 = g0 + (size_t)BB * LL * DD;

    dim3 grid1(DD / 16, LL / 16, BB);   // 25 x 16 x 8 tiles, 1 wave each
    relation_gemm_wmma<<<grid1, 32, 0, stream>>>(x, W0, b0v, W1, b1v, g0, g1);

    dim3 grid2(DD, BB);                 // 3200 blocks x 256 threads
    relation_gate_kernel<<<grid2, LL, 0, stream>>>(x, mask, g0, g1, out);
}